// SuperLoss_8083128451690
// MI455X (gfx1250) — compile-verified
//
#include <hip/hip_runtime.h>
#include <hip/hip_bf16.h>
#include <stdint.h>

// ---------------- SuperLoss constants ----------------
#define TAU_F        1.5f
#define INV_LAM_F    (1.0f / 0.9f)
#define LAM_F        0.9f
#define E_F          2.71828182845904523536f
#define NEG2_OVER_E  (-0.73575888234288464320f)
#define HALLEY_ITERS 4
#define INV_BATCH    (1.0f / 32.0f)

// ---------------- launch geometry --------------------
#define BLOCK   256          // 8 wave32 waves
#define VEC     4            // float4 per thread per chunk
#define CHUNK   (BLOCK*VEC)  // 1024 f32 per array per block-iteration (4 KB)
#define MAXGRID 2048

// ---------------- CDNA5 async-copy helpers -----------
// Per-lane async DMA of 16B global -> LDS, tracked by ASYNCcnt (not LOADcnt).
// Inline asm per CDNA5 ISA — portable across toolchains.
__device__ __forceinline__ void async_g2lds_b128(const float* gaddr, unsigned lds_byte_off) {
    asm volatile("global_load_async_to_lds_b128 %0, %1, off"
                 :: "v"(lds_byte_off), "v"(gaddr) : "memory");
}
__device__ __forceinline__ void wait_async_le2() {
    asm volatile("s_wait_asynccnt 0x2" ::: "memory");
}
__device__ __forceinline__ void wait_async_le0() {
    asm volatile("s_wait_asynccnt 0x0" ::: "memory");
}

// On AMDGPU the low 32 bits of a flat pointer into LDS are the workgroup-
// relative LDS byte address (the shared aperture lives in the high bits).
__device__ __forceinline__ unsigned lds_off(const void* p) {
    return (unsigned)(uintptr_t)p;
}

// ---------------- per-element SuperLoss --------------
__device__ __forceinline__ float superloss_elem(float p, float t) {
    // BCE with logs clamped at -100 (torch BCELoss semantics).
    // log1p(-p) == log(1-p): 1-p is exact in f32 for p>=0.5 (Sterbenz) and
    // perturbs log by <= ~1.2e-7 absolute for p<0.5 — hardware v_log path.
    float logp  = fmaxf(__logf(p), -100.0f);
    float log1m = fmaxf(__logf(1.0f - p), -100.0f);
    float l = -(t * logp + (1.0f - t) * log1m);

    // y = 0.5*max(-2/e, (l - tau)/lam)  >= -1/e
    float y = 0.5f * fmaxf(NEG2_OVER_E, (l - TAU_F) * INV_LAM_F);

    // Lambert W init: series near branch point for y<0, log(1+y) elsewhere.
    // (Init only — Halley's cubic convergence erases O(1e-7) init error.)
    float w_neg = -1.0f + __builtin_sqrtf(fmaxf(2.0f * __builtin_fmaf(E_F, y, 1.0f), 0.0f));
    float w_pos = __logf(1.0f + y);
    float w = (y < 0.0f) ? w_neg : w_pos;

    // Halley, cubically convergent; 4 iters ≈ f32-converged from this init.
    // Rearranged to a single v_rcp per iteration:
    //   w -= 2a*f / (2a^2*e^w - (w+2)*f),  a = w+1,  f = w*e^w - y
#pragma unroll
    for (int i = 0; i < HALLEY_ITERS; ++i) {
        float ew  = __expf(w);
        float a   = w + 1.0f;
        float f   = __builtin_fmaf(w, ew, -y);
        float num = 2.0f * a * f;
        float den = __builtin_fmaf(2.0f * a * a, ew, -(w + 2.0f) * f) + 1e-12f;
        w = __builtin_fmaf(-num, __builtin_amdgcn_rcpf(den), w);
    }

    float sigma = __expf(-w);                       // exp(-W(y))
    return __builtin_fmaf(l - TAU_F, sigma, LAM_F * w * w);
}

// ---------------- block reduction --------------------
__device__ __forceinline__ float block_reduce_sum(float v) {
    // wave32 butterfly
#pragma unroll
    for (int off = 16; off > 0; off >>= 1)
        v += __shfl_xor(v, off, 32);
    __shared__ float swave[BLOCK / 32];
    const int tid = threadIdx.x;
    if ((tid & 31) == 0) swave[tid >> 5] = v;
    __syncthreads();
    float r = 0.0f;
    if (tid < (BLOCK / 32)) {
        r = swave[tid];
#pragma unroll
        for (int off = (BLOCK / 64); off > 0; off >>= 1)
            r += __shfl_xor(r, off, 32);
    }
    return r;   // valid on tid==0
}

// ---------------- stage 1: stream + partial sums -----
__global__ __launch_bounds__(BLOCK) void superloss_stage1(
    const float* __restrict__ logits, const float* __restrict__ targets,
    float* __restrict__ partials, long long full_chunks, long long N)
{
    __shared__ float sL[2][CHUNK];
    __shared__ float sT[2][CHUNK];

    const int tid = threadIdx.x;
    const unsigned elt = (unsigned)tid * VEC;

    const unsigned loL[2] = { lds_off(&sL[0][elt]), lds_off(&sL[1][elt]) };
    const unsigned loT[2] = { lds_off(&sT[0][elt]), lds_off(&sT[1][elt]) };

    float acc = 0.0f;

    // double-buffered async pipeline over full chunks
    long long c0 = blockIdx.x;
    if (c0 < full_chunks) {
        async_g2lds_b128(logits  + c0 * CHUNK + elt, loL[0]);
        async_g2lds_b128(targets + c0 * CHUNK + elt, loT[0]);
    }
    int buf = 0;
    for (long long c = c0; c < full_chunks; c += gridDim.x) {
        long long cn = c + gridDim.x;
        if (cn < full_chunks) {
            async_g2lds_b128(logits  + cn * CHUNK + elt, loL[buf ^ 1]);
            async_g2lds_b128(targets + cn * CHUNK + elt, loT[buf ^ 1]);
            wait_async_le2();       // current chunk's 2 oldest ops complete
        } else {
            wait_async_le0();
        }
        float4 pv = *(const float4*)&sL[buf][elt];   // ds_load_b128
        float4 tv = *(const float4*)&sT[buf][elt];
        acc += superloss_elem(pv.x, tv.x);
        acc += superloss_elem(pv.y, tv.y);
        acc += superloss_elem(pv.z, tv.z);
        acc += superloss_elem(pv.w, tv.w);
        buf ^= 1;
    }

    // tail (N % CHUNK) handled by block 0 with direct loads
    if (blockIdx.x == 0) {
        for (long long i = full_chunks * CHUNK + tid; i < N; i += BLOCK)
            acc += superloss_elem(logits[i], targets[i]);
    }

    float total = block_reduce_sum(acc);
    if (tid == 0) partials[blockIdx.x] = total;
}

// ---------------- stage 2: deterministic final reduce
__global__ __launch_bounds__(BLOCK) void superloss_stage2(
    const float* __restrict__ partials, int n, float* __restrict__ out)
{
    float acc = 0.0f;
    for (int i = threadIdx.x; i < n; i += BLOCK)
        acc += partials[i];
    float total = block_reduce_sum(acc);
    if (threadIdx.x == 0) out[0] = total * INV_BATCH;
}

// ---------------- host entry -------------------------
extern "C" void kernel_launch(void* const* d_in, const int* in_sizes, int n_in,
                              void* d_out, int out_size, void* d_ws, size_t ws_size,
                              hipStream_t stream) {
    const float* logits  = (const float*)d_in[0];
    const float* targets = (const float*)d_in[1];
    float* out      = (float*)d_out;
    float* partials = (float*)d_ws;

    long long N = (long long)in_sizes[0];
    long long full_chunks = N / CHUNK;

    long long grid = MAXGRID;
    if (grid > full_chunks) grid = full_chunks;
    long long ws_cap = (long long)(ws_size / sizeof(float));
    if (grid > ws_cap) grid = ws_cap;
    if (grid < 1) grid = 1;

    superloss_stage1<<<(dim3)(unsigned)grid, BLOCK, 0, stream>>>(
        logits, targets, partials, full_chunks, N);
    superloss_stage2<<<1, BLOCK, 0, stream>>>(partials, (int)grid, out);
}